// CoPE_76759655514280
// MI455X (gfx1250) — compile-verified
//
#include <hip/hip_runtime.h>
#include <hip/hip_bf16.h>

typedef float v2f __attribute__((ext_vector_type(2)));
typedef float v8f __attribute__((ext_vector_type(8)));

#define SEQ 2048
#define NHEAD 16
#define DIM 64
#define NPOS 64

__device__ __forceinline__ float sigmoidf_(float x) {
  return 1.0f / (1.0f + __expf(-x));
}

// Gather + lerp from the per-row 64-entry position-logit table.
__device__ __forceinline__ float cope_interp(float pos, const float* __restrict__ tab) {
  pos = fminf(pos, 63.0f);            // pos > 0 always (suffix sum of sigmoids)
  float pf = floorf(pos);
  float w  = pos - pf;
  int   fi = (int)pf;
  int   ci = (int)ceilf(pos);
  return tab[ci] * w + tab[fi] * (1.0f - w);
}

// One block = one (head h, 16 query rows m0..m0+15) tile.
// 128 threads = 4 waves (wave32).
__global__ __launch_bounds__(128) void cope_kernel(
    const float* __restrict__ q,     // [H, S, D]
    const float* __restrict__ attn,  // [H, S, S]
    const float* __restrict__ pe,    // [D, NPOS]
    float* __restrict__ out) {       // [H, S, S]
  __shared__ float s_table[16 * NPOS];   // logits_int tile: [16 rows][64 positions]
  __shared__ float s_excl[4][SEQ];       // per-wave exclusive-prefix staging (8 KB each)

  const int h    = blockIdx.y;
  const int m0   = blockIdx.x * 16;
  const int lane = threadIdx.x & 31;
  const int wv   = threadIdx.x >> 5;     // 0..3

  // ---------------- Phase 1: logits_int tile via V_WMMA_F32_16X16X4_F32 -------------
  // Wave wv computes the 16x16 output tile at columns p0 = 16*wv, accumulating K=64
  // in 16 steps of K=4. A: 16x4 f32 layout (lanes 0-15: K=k0,k0+1; lanes 16-31: K=k0+2,k0+3).
  // B (4x16): symmetric layout with lane = N column.
  {
    const int p0   = wv * 16;
    const int half = lane >> 4;          // 0 or 1
    const int l    = lane & 15;
    const float* Q = q + ((size_t)(h * SEQ + m0)) * DIM;   // 16 x 64 row-major
    v8f acc = {};
#pragma unroll
    for (int kk = 0; kk < 16; ++kk) {
      const int k0 = kk * 4 + half * 2;
      v2f a, b;
      a.x = Q[l * DIM + k0];
      a.y = Q[l * DIM + k0 + 1];
      b.x = pe[(k0)     * NPOS + p0 + l];
      b.y = pe[(k0 + 1) * NPOS + p0 + l];
      acc = __builtin_amdgcn_wmma_f32_16x16x4_f32(
          /*neg_a=*/false, a, /*neg_b=*/false, b,
          /*c_mod=*/(short)0, acc, /*reuse_a=*/false, /*reuse_b=*/false);
    }
    // C/D layout: VGPR r -> row M = r + 8*half ; lane (mod 16) -> column N.
    const int mbase = half * 8;
#pragma unroll
    for (int r = 0; r < 8; ++r)
      s_table[(mbase + r) * NPOS + p0 + l] = acc[r];
  }
  __syncthreads();

  // ---------------- Phase 2: suffix-scan + gather/lerp, 4 rows per wave --------------
  float* ex = s_excl[wv];
  for (int rr = 0; rr < 4; ++rr) {
    const int lrow = wv * 4 + rr;        // 0..15
    const int s    = m0 + lrow;
    const float* arow = attn + ((size_t)(h * SEQ + s)) * SEQ;
    float*       orow = out  + ((size_t)(h * SEQ + s)) * SEQ;
    const float* tab  = s_table + lrow * NPOS;

    // Pass 1: forward exclusive prefix of sigmoid(gates) into LDS; accumulate total.
    float running = 0.0f;
#pragma unroll 1
    for (int c = 0; c < SEQ / 128; ++c) {
      const int base = c * 128 + lane * 4;           // coalesced 512B per wave
      float4 x = *(const float4*)(arow + base);
      float g0 = sigmoidf_(x.x), g1 = sigmoidf_(x.y);
      float g2 = sigmoidf_(x.z), g3 = sigmoidf_(x.w);
      float lsum = (g0 + g1) + (g2 + g3);
      float incl = lsum;
#pragma unroll
      for (int off = 1; off < 32; off <<= 1) {
        float t = __shfl_up(incl, off, 32);
        if (lane >= off) incl += t;
      }
      float lex = running + (incl - lsum);           // exclusive prefix for element 0
      ex[base + 0] = lex;
      ex[base + 1] = lex + g0;
      ex[base + 2] = lex + g0 + g1;
      ex[base + 3] = lex + g0 + g1 + g2;
      running += __shfl(incl, 31, 32);               // super-chunk total carry
    }
    const float total = running;                     // row total => reverse cumsum base

    // Pass 2: pos = min(total - excl, 63); gather floor/ceil from table; lerp; store.
#pragma unroll 1
    for (int c = 0; c < SEQ / 128; ++c) {
      const int base = c * 128 + lane * 4;
      float4 e = *(const float4*)(ex + base);
      float4 o;
      o.x = cope_interp(total - e.x, tab);
      o.y = cope_interp(total - e.y, tab);
      o.z = cope_interp(total - e.z, tab);
      o.w = cope_interp(total - e.w, tab);
      *(float4*)(orow + base) = o;                   // coalesced 512B per wave
    }
  }
}

extern "C" void kernel_launch(void* const* d_in, const int* in_sizes, int n_in,
                              void* d_out, int out_size, void* d_ws, size_t ws_size,
                              hipStream_t stream) {
  (void)in_sizes; (void)n_in; (void)d_ws; (void)ws_size; (void)out_size;
  const float* q    = (const float*)d_in[0];   // [1,16,2048,64]
  const float* attn = (const float*)d_in[1];   // [1,16,2048,2048]
  const float* pe   = (const float*)d_in[2];   // [64,64]
  float* out = (float*)d_out;                  // [1,16,2048,2048]

  dim3 grid(SEQ / 16, NHEAD, 1);               // 128 x 16 = 2048 blocks
  cope_kernel<<<grid, 128, 0, stream>>>(q, attn, pe, out);
}